// MHA_14431090115192
// MI455X (gfx1250) — compile-verified
//
#include <hip/hip_runtime.h>
#include <math.h>

typedef __attribute__((ext_vector_type(2))) float v2f;
typedef __attribute__((ext_vector_type(8))) float v8f;

#define E 128
#define NSEQ 512
#define BATCH 64
#define HEADS 8
#define HD 16
#define ROWS (BATCH * NSEQ)          // 32768
#define FF 512
#define SLD 516                      // LDS pitch (floats): 4*r mod 64 distinct for r=0..15
#define EPS 1e-5f

// ---------------- WMMA helpers (V_WMMA_F32_16X16X4_F32) ----------------
__device__ __forceinline__ v8f wmma4(v2f a, v2f b, v8f c) {
  return __builtin_amdgcn_wmma_f32_16x16x4_f32(
      /*neg_a=*/false, a, /*neg_b=*/false, b,
      /*c_mod=*/(short)0, c, /*reuse_a=*/false, /*reuse_b=*/false);
}

// A fragment: 16x4 tile at `A` (row-major, leading dim lda).
// lanes 0-15: M=lane, K=0 (x) / K=1 (y); lanes 16-31: M=lane-16, K=2/3.
__device__ __forceinline__ v2f load_a(const float* A, int lda, int lane) {
  int row = lane & 15;
  int k = (lane >> 4) << 1;
  v2f a;
  a.x = A[row * lda + k];
  a.y = A[row * lda + k + 1];
  return a;
}

// B fragment: 4x16 tile at `Bp` (row-major, leading dim ldb).
// lanes 0-15: N=lane, K=0 (x) / 1 (y); lanes 16-31: N=lane-16, K=2/3.
__device__ __forceinline__ v2f load_b(const float* Bp, int ldb, int lane) {
  int col = lane & 15;
  int k = (lane >> 4) << 1;
  v2f b;
  b.x = Bp[k * ldb + col];
  b.y = Bp[(k + 1) * ldb + col];
  return b;
}

// ---------------- Kernel 1: fused Q/K/V projections ----------------
// grid = ROWS/16 blocks, 256 threads (8 waves). Wave w owns output col-tile w
// for all three matrices (shares the A fragment across 3 WMMA chains).
__global__ void qkv_proj(const float* __restrict__ X,
                         const float* __restrict__ Wq, const float* __restrict__ bq,
                         const float* __restrict__ Wk, const float* __restrict__ bk,
                         const float* __restrict__ Wv, const float* __restrict__ bv,
                         float* __restrict__ Q, float* __restrict__ K,
                         float* __restrict__ V) {
  int wave = threadIdx.x >> 5, lane = threadIdx.x & 31;
  int m0 = blockIdx.x * 16;
  int n0 = wave * 16;
  const float* Ap = X + (size_t)m0 * E;
  v8f cq = {}, ck = {}, cv = {};
  for (int k0 = 0; k0 < E; k0 += 4) {
    v2f a = load_a(Ap + k0, E, lane);
    v2f fq = load_b(Wq + (size_t)k0 * E + n0, E, lane);
    v2f fk = load_b(Wk + (size_t)k0 * E + n0, E, lane);
    v2f fv = load_b(Wv + (size_t)k0 * E + n0, E, lane);
    cq = wmma4(a, fq, cq);
    ck = wmma4(a, fk, ck);
    cv = wmma4(a, fv, cv);
  }
  int col = n0 + (lane & 15);
  int rb = m0 + ((lane >> 4) << 3);
  for (int r = 0; r < 8; ++r) {
    size_t off = (size_t)(rb + r) * E + col;
    Q[off] = cq[r] + bq[col];
    K[off] = ck[r] + bk[col];
    V[off] = cv[r] + bv[col];
  }
}

// ---------------- Kernel 2: attention per (b, head, 16-query tile) ----------
// 1 wave per block; 16x512 score slab in LDS; softmax via shfl reductions.
__global__ void attention(const float* __restrict__ Q, const float* __restrict__ K,
                          const float* __restrict__ V, float* __restrict__ O) {
  __shared__ float S[16 * SLD];
  int lane = threadIdx.x;
  int idx = blockIdx.x;                 // ((b*HEADS)+m)*32 + it
  int it = idx & 31; idx >>= 5;
  int m = idx & 7;
  int b = idx >> 3;
  const float* Qp = Q + ((size_t)b * NSEQ + it * 16) * E + m * HD;
  const float* Kb = K + (size_t)b * NSEQ * E + m * HD;
  const float* Vb = V + (size_t)b * NSEQ * E + m * HD;

  // scores: S[i, j] = (Q_tile . K_j) * 1/sqrt(16)
  for (int jt = 0; jt < 32; ++jt) {
    const float* Kp = Kb + (size_t)jt * 16 * E;
    v8f c = {};
    for (int k0 = 0; k0 < HD; k0 += 4) {
      v2f a = load_a(Qp + k0, E, lane);
      int colj = lane & 15;
      int kk = k0 + ((lane >> 4) << 1);
      v2f bf;                            // B = K^T tile: B[k][j] = K[j][k]
      bf.x = Kp[colj * E + kk];
      bf.y = Kp[colj * E + kk + 1];
      c = wmma4(a, bf, c);
    }
    int col = jt * 16 + (lane & 15);
    int rb = (lane >> 4) << 3;
    for (int r = 0; r < 8; ++r) S[(rb + r) * SLD + col] = c[r] * 0.25f;
  }
  __syncthreads();

  // softmax over j (512) per row, wave32 reductions
  for (int r = 0; r < 16; ++r) {
    float mx = -INFINITY;
    for (int j = lane; j < NSEQ; j += 32) mx = fmaxf(mx, S[r * SLD + j]);
    for (int off = 16; off; off >>= 1) mx = fmaxf(mx, __shfl_xor(mx, off, 32));
    float sum = 0.f;
    for (int j = lane; j < NSEQ; j += 32) {
      float e = __expf(S[r * SLD + j] - mx);
      S[r * SLD + j] = e;
      sum += e;
    }
    for (int off = 16; off; off >>= 1) sum += __shfl_xor(sum, off, 32);
    float inv = 1.0f / sum;
    for (int j = lane; j < NSEQ; j += 32) S[r * SLD + j] *= inv;
  }
  __syncthreads();

  // out = P(16x512) @ V(512x16)
  v8f c = {};
  for (int k0 = 0; k0 < NSEQ; k0 += 4) {
    v2f a = load_a(&S[k0], SLD, lane);
    v2f bf = load_b(Vb + (size_t)k0 * E, E, lane);
    c = wmma4(a, bf, c);
  }
  float* Op = O + ((size_t)b * NSEQ + it * 16) * E + m * HD;
  int col = lane & 15;
  int rb = (lane >> 4) << 3;
  for (int r = 0; r < 8; ++r) Op[(rb + r) * E + col] = c[r];
}

// ---------------- Kernel 3: output projection + residual ----------------
__global__ void out_proj(const float* __restrict__ A, const float* __restrict__ W,
                         const float* __restrict__ bias, const float* __restrict__ R,
                         float* __restrict__ Y) {
  int wave = threadIdx.x >> 5, lane = threadIdx.x & 31;
  int m0 = blockIdx.x * 16;
  int n0 = wave * 16;
  const float* Ap = A + (size_t)m0 * E;
  v8f c = {};
  for (int k0 = 0; k0 < E; k0 += 4) {
    v2f a = load_a(Ap + k0, E, lane);
    v2f bf = load_b(W + (size_t)k0 * E + n0, E, lane);
    c = wmma4(a, bf, c);
  }
  int col = n0 + (lane & 15);
  int rb = m0 + ((lane >> 4) << 3);
  for (int r = 0; r < 8; ++r) {
    size_t off = (size_t)(rb + r) * E + col;
    Y[off] = c[r] + bias[col] + R[off];
  }
}

// ---------------- Kernel 4: fused FFN (128 -> 512 ReLU -> 128) + residual ---
__global__ void ffn(const float* __restrict__ X,
                    const float* __restrict__ W1, const float* __restrict__ b1,
                    const float* __restrict__ W2, const float* __restrict__ b2,
                    float* __restrict__ Y) {
  __shared__ float H[16 * SLD];
  int wave = threadIdx.x >> 5, lane = threadIdx.x & 31;
  int m0 = blockIdx.x * 16;
  const float* Ap = X + (size_t)m0 * E;

  // phase 1: H = relu(X @ W1 + b1), H is 16x512 in LDS
  for (int t = wave; t < 32; t += 8) {
    int n0 = t * 16;
    v8f c = {};
    for (int k0 = 0; k0 < E; k0 += 4) {
      v2f a = load_a(Ap + k0, E, lane);
      v2f bf = load_b(W1 + (size_t)k0 * FF + n0, FF, lane);
      c = wmma4(a, bf, c);
    }
    int col = n0 + (lane & 15);
    int rb = (lane >> 4) << 3;
    for (int r = 0; r < 8; ++r)
      H[(rb + r) * SLD + col] = fmaxf(c[r] + b1[col], 0.0f);
  }
  __syncthreads();

  // phase 2: Y = H @ W2 + b2 + X
  int n0 = wave * 16;
  v8f c = {};
  for (int k0 = 0; k0 < FF; k0 += 4) {
    v2f a = load_a(&H[k0], SLD, lane);
    v2f bf = load_b(W2 + (size_t)k0 * E + n0, E, lane);
    c = wmma4(a, bf, c);
  }
  int col = n0 + (lane & 15);
  int rb = (lane >> 4) << 3;
  for (int r = 0; r < 8; ++r) {
    size_t off = (size_t)(m0 + rb + r) * E + col;
    Y[off] = c[r] + b2[col] + X[off];
  }
}

// ---------------- BatchNorm: deterministic two-stage reduction --------------
// Stage 1: 256 blocks x 256 threads; block covers 128 rows; writes per-block
// partial sum/sumsq per channel.
__global__ void bn_stats1(const float* __restrict__ X, float* __restrict__ part) {
  __shared__ float ls[256], lq[256];
  int t = threadIdx.x;
  int ch = t & 127;
  int rl = t >> 7;                       // 0 or 1
  int r0 = blockIdx.x * 128 + rl;
  float s = 0.f, q = 0.f;
  for (int i = 0; i < 64; ++i) {
    float v = X[(size_t)(r0 + 2 * i) * E + ch];
    s += v;
    q += v * v;
  }
  ls[t] = s; lq[t] = q;
  __syncthreads();
  if (t < 128) {
    float ss = ls[t] + ls[t + 128];
    float qq = lq[t] + lq[t + 128];
    part[(size_t)blockIdx.x * 256 + ch] = ss;
    part[(size_t)blockIdx.x * 256 + 128 + ch] = qq;
  }
}

// Stage 2: 1 block, 128 threads: combine 256 partials per channel.
__global__ void bn_stats2(const float* __restrict__ part, float* __restrict__ stat) {
  int ch = threadIdx.x;
  float s = 0.f, q = 0.f;
  for (int blk = 0; blk < 256; ++blk) {
    s += part[(size_t)blk * 256 + ch];
    q += part[(size_t)blk * 256 + 128 + ch];
  }
  stat[ch] = s;
  stat[128 + ch] = q;
}

__global__ void bn_apply(const float* __restrict__ X, const float* __restrict__ stat,
                         const float* __restrict__ g, const float* __restrict__ bta,
                         float* __restrict__ Y) {
  size_t i = (size_t)blockIdx.x * blockDim.x + threadIdx.x;
  int ch = (int)(i & 127);
  float mu = stat[ch] * (1.0f / (float)ROWS);
  float var = stat[128 + ch] * (1.0f / (float)ROWS) - mu * mu;
  float sc = rsqrtf(var + EPS) * g[ch];
  Y[i] = (X[i] - mu) * sc + bta[ch];
}

// ---------------- launch ----------------
extern "C" void kernel_launch(void* const* d_in, const int* in_sizes, int n_in,
                              void* d_out, int out_size, void* d_ws, size_t ws_size,
                              hipStream_t stream) {
  const float* x0  = (const float*)d_in[0];   // embedding_node (B,N,E)
  const float* wq  = (const float*)d_in[1];
  const float* bq  = (const float*)d_in[2];
  const float* wk  = (const float*)d_in[3];
  const float* bk  = (const float*)d_in[4];
  const float* wv  = (const float*)d_in[5];
  const float* bv  = (const float*)d_in[6];
  const float* ww  = (const float*)d_in[7];
  const float* wb  = (const float*)d_in[8];
  const float* f1w = (const float*)d_in[9];
  const float* f1b = (const float*)d_in[10];
  const float* f2w = (const float*)d_in[11];
  const float* f2b = (const float*)d_in[12];
  const float* g1  = (const float*)d_in[13];
  const float* be1 = (const float*)d_in[14];
  const float* g2  = (const float*)d_in[15];
  const float* be2 = (const float*)d_in[16];

  const size_t MAT = (size_t)ROWS * E;        // 4,194,304 floats
  float* ws   = (float*)d_ws;
  float* Q    = ws;
  float* K    = ws + MAT;
  float* V    = ws + 2 * MAT;
  float* ATT  = ws + 3 * MAT;
  float* X1   = ws + 4 * MAT;
  float* X2   = ws + 5 * MAT;
  float* PART = ws + 6 * MAT;                 // 256*256 floats
  float* STAT = PART + 256 * 256;             // 256 floats
  float* OUT  = (float*)d_out;

  const int rowTiles = ROWS / 16;             // 2048

  qkv_proj<<<rowTiles, 256, 0, stream>>>(x0, wq, bq, wk, bk, wv, bv, Q, K, V);
  attention<<<BATCH * HEADS * (NSEQ / 16), 32, 0, stream>>>(Q, K, V, ATT);
  out_proj<<<rowTiles, 256, 0, stream>>>(ATT, ww, wb, x0, X1);

  bn_stats1<<<256, 256, 0, stream>>>(X1, PART);
  bn_stats2<<<1, 128, 0, stream>>>(PART, STAT);
  bn_apply<<<(int)(MAT / 256), 256, 0, stream>>>(X1, STAT, g1, be1, X1);

  ffn<<<rowTiles, 256, 0, stream>>>(X1, f1w, f1b, f2w, f2b, X2);

  bn_stats1<<<256, 256, 0, stream>>>(X2, PART);
  bn_stats2<<<1, 128, 0, stream>>>(PART, STAT);
  bn_apply<<<(int)(MAT / 256), 256, 0, stream>>>(X2, STAT, g2, be2, OUT);
}